// StructureDecoder_23871428231492
// MI455X (gfx1250) — compile-verified
//
#include <hip/hip_runtime.h>

#define N_NODES 10000
#define N_EDGES 320000
#define DIM     64
#define MT      625           // N_NODES / 16 tiles per dimension
#define TN      5             // n-tiles per wave in the Gram GEMM (625 = 5*125)
#define NG      (MT / TN)     // 125 n-tile groups

typedef __attribute__((ext_vector_type(2))) float v2f;
typedef __attribute__((ext_vector_type(8))) float v8f;

// D = A(16x4) * B(4x16) + C, f32 everywhere.
// A layout (ISA 7.12.2): lanes 0-15 -> M=lane, K={0,1} in the 2 VGPRs;
//                        lanes 16-31 -> M=lane-16, K={2,3}.
// B layout mirrors: lanes 0-15 -> N=lane, K={0,1}; lanes 16-31 -> N=lane-16, K={2,3}.
__device__ __forceinline__ v8f wmma4(v2f a, v2f b, v8f c) {
  return __builtin_amdgcn_wmma_f32_16x16x4_f32(
      /*neg_a=*/false, a, /*neg_b=*/false, b,
      /*c_mod=*/(short)0, c, /*reuse_a=*/false, /*reuse_b=*/false);
}

// --- 1) init: deg = 1 (self loop), agg = 0 ---------------------------------
__global__ __launch_bounds__(256) void k_init(float* __restrict__ deg,
                                              float* __restrict__ agg) {
  int i = blockIdx.x * blockDim.x + threadIdx.x;
  if (i < N_NODES * DIM) agg[i] = 0.0f;
  if (i < N_NODES)       deg[i] = 1.0f;
}

// --- 2) degree count over destination nodes --------------------------------
__global__ __launch_bounds__(256) void k_deg(const long long* __restrict__ col,
                                             float* __restrict__ deg) {
  int e = blockIdx.x * blockDim.x + threadIdx.x;
  if (e < N_EDGES) atomicAdd(&deg[(int)col[e]], 1.0f);
}

// --- 3) deg -> rsqrt(deg), in place ----------------------------------------
__global__ __launch_bounds__(256) void k_dinv(float* __restrict__ deg) {
  int i = blockIdx.x * blockDim.x + threadIdx.x;
  if (i < N_NODES) {
    float d = deg[i];
    deg[i] = (d > 0.0f) ? rsqrtf(d) : 0.0f;
  }
}

// --- 4) h = z @ W  (WMMA f32 16x16x4) --------------------------------------
__global__ __launch_bounds__(256) void k_h(const float* __restrict__ z,
                                           const float* __restrict__ W,
                                           float* __restrict__ h) {
  int wave = blockIdx.x * (blockDim.x >> 5) + (threadIdx.x >> 5);
  const int ntile = MT * (DIM / 16);            // 625 * 4 = 2500 tiles
  if (wave >= ntile) return;                    // wave-uniform: EXEC stays all-1s
  int lane = threadIdx.x & 31;
  int tm = wave >> 2;
  int tn = wave & 3;
  int l15  = lane & 15;
  int koff = (lane >> 4) << 1;                  // 0 or 2
  int mrow = tm * 16 + l15;
  int ncol = tn * 16 + l15;
  const float* za = z + mrow * DIM + koff;
  v8f acc = {};
#pragma unroll
  for (int k = 0; k < DIM; k += 4) {
    v2f a = { za[k], za[k + 1] };
    v2f b = { W[(k + koff) * DIM + ncol], W[(k + koff + 1) * DIM + ncol] };
    acc = wmma4(a, b, acc);
  }
  int rbase = tm * 16 + ((lane >> 4) << 3);
#pragma unroll
  for (int v = 0; v < 8; ++v)
    h[(rbase + v) * DIM + ncol] = acc[v];
}

// --- 5) edge scatter: agg[c] += h[r] * dinv[r]*dinv[c] ---------------------
// 16 threads per edge, each handles 4 consecutive features (float4 load).
__global__ __launch_bounds__(256) void k_scatter(const long long* __restrict__ row,
                                                 const long long* __restrict__ col,
                                                 const float* __restrict__ h,
                                                 const float* __restrict__ dinv,
                                                 float* __restrict__ agg) {
  int t = blockIdx.x * blockDim.x + threadIdx.x;
  if (t >= N_EDGES * 16) return;
  int e  = t >> 4;
  int k0 = (t & 15) << 2;
  int r = (int)row[e];
  int c = (int)col[e];
  float norm = dinv[r] * dinv[c];
  float4 hv = *(const float4*)(h + r * DIM + k0);
  float* a = agg + c * DIM + k0;
  atomicAdd(a + 0, hv.x * norm);
  atomicAdd(a + 1, hv.y * norm);
  atomicAdd(a + 2, hv.z * norm);
  atomicAdd(a + 3, hv.w * norm);
}

// --- 6) self-loop term + bias + ReLU, in place in agg ----------------------
__global__ __launch_bounds__(256) void k_finalize(const float* __restrict__ h,
                                                  const float* __restrict__ dinv,
                                                  const float* __restrict__ bias,
                                                  float* __restrict__ agg) {
  int i = blockIdx.x * blockDim.x + threadIdx.x;
  if (i >= N_NODES * DIM) return;
  int node = i >> 6;
  int k    = i & 63;
  float di = dinv[node];
  float v  = agg[i] + h[i] * di * di + bias[k];
  agg[i] = fmaxf(v, 0.0f);
}

// --- 7) out = zr @ zr.T  (WMMA f32 16x16x4, 5 n-tiles per wave) ------------
__global__ __launch_bounds__(256) void k_gram(const float* __restrict__ zr,
                                              float* __restrict__ out) {
  int wave = blockIdx.x * (blockDim.x >> 5) + (threadIdx.x >> 5);
  const int nwaves = MT * NG;                   // 625 * 125 = 78125
  if (wave >= nwaves) return;                   // wave-uniform guard
  int lane = threadIdx.x & 31;
  int tm  = wave / NG;
  int tn0 = (wave % NG) * TN;
  int l15  = lane & 15;
  int koff = (lane >> 4) << 1;                  // 0 or 2
  // A tile rows (m) and B tile rows (n): B[k,n] = zr[n,k], identical layout.
  const float* za = zr + (tm  * 16 + l15) * DIM + koff;
  const float* zb = zr + (tn0 * 16 + l15) * DIM + koff;
  v8f acc[TN] = {};
#pragma unroll
  for (int k = 0; k < DIM; k += 4) {
    v2f a = { za[k], za[k + 1] };
#pragma unroll
    for (int t = 0; t < TN; ++t) {
      const float* p = zb + t * (16 * DIM);
      v2f b = { p[k], p[k + 1] };
      acc[t] = wmma4(a, b, acc[t]);
    }
  }
  int rbase = tm * 16 + ((lane >> 4) << 3);
#pragma unroll
  for (int t = 0; t < TN; ++t) {
    size_t cbase = (size_t)(tn0 + t) * 16 + l15;
#pragma unroll
    for (int v = 0; v < 8; ++v)
      out[(size_t)(rbase + v) * N_NODES + cbase] = acc[t][v];
  }
}

extern "C" void kernel_launch(void* const* d_in, const int* in_sizes, int n_in,
                              void* d_out, int out_size, void* d_ws, size_t ws_size,
                              hipStream_t stream) {
  const float*     z    = (const float*)d_in[0];       // [N, 64]
  const long long* ei   = (const long long*)d_in[1];   // [2, E] int64
  const float*     W    = (const float*)d_in[2];       // [64, 64]
  const float*     bias = (const float*)d_in[3];       // [64]
  float*           out  = (float*)d_out;               // [N, N]

  const long long* row = ei;                 // source nodes
  const long long* col = ei + N_EDGES;       // destination nodes

  // workspace layout (floats): deg/dinv [N] | h [N*64] | agg/zr [N*64]
  float* deg = (float*)d_ws;
  float* h   = deg + N_NODES;
  float* agg = h + (size_t)N_NODES * DIM;

  k_init    <<<(N_NODES * DIM + 255) / 256, 256, 0, stream>>>(deg, agg);
  k_deg     <<<(N_EDGES + 255) / 256,       256, 0, stream>>>(col, deg);
  k_dinv    <<<(N_NODES + 255) / 256,       256, 0, stream>>>(deg);
  k_h       <<<(MT * 4 + 7) / 8,            256, 0, stream>>>(z, W, h);
  k_scatter <<<(N_EDGES * 16 + 255) / 256,  256, 0, stream>>>(row, col, h, deg, agg);
  k_finalize<<<(N_NODES * DIM + 255) / 256, 256, 0, stream>>>(h, deg, bias, agg);
  k_gram    <<<(MT * NG + 7) / 8,           256, 0, stream>>>(agg, out);
}